// GCNMovieRecommender_25065429139628
// MI455X (gfx1250) — compile-verified
//
#include <hip/hip_runtime.h>

#define NNODES 50000
#define NEDGES 800000

typedef float v2f __attribute__((ext_vector_type(2)));
typedef float v8f __attribute__((ext_vector_type(8)));

// ---------------- small utility kernels ----------------

__global__ void zero_f32(float* __restrict__ p, int n) {
    int i = blockIdx.x * blockDim.x + threadIdx.x;
    if (i < n) p[i] = 0.0f;
}

__global__ void deg_count(const int* __restrict__ dst, float* __restrict__ deg, int E) {
    int i = blockIdx.x * blockDim.x + threadIdx.x;
    if (i < E) atomicAdd(&deg[dst[i]], 1.0f);
}

// self-loop contributes +1 to every node's degree -> deg is always > 0
__global__ void deg_to_dinv(float* __restrict__ d, int n) {
    int i = blockIdx.x * blockDim.x + threadIdx.x;
    if (i < n) d[i] = rsqrtf(d[i] + 1.0f);
}

__global__ void edge_norm(const int* __restrict__ src, const int* __restrict__ dst,
                          const float* __restrict__ dinv, float* __restrict__ norm, int E) {
    int i = blockIdx.x * blockDim.x + threadIdx.x;
    if (i < E) norm[i] = dinv[src[i]] * dinv[dst[i]];
}

// out[i*4 .. i*4+3] = broadcast bias, float4-vectorized. F4 = F/4 (power of two).
__global__ void bias_init4(float4* __restrict__ out, const float4* __restrict__ b,
                           int total4, int F4mask) {
    int i = blockIdx.x * blockDim.x + threadIdx.x;
    if (i < total4) out[i] = b[i & F4mask];
}

// ------------- fp32 WMMA GEMM: out[M,F] = (relu?)A[M,K] @ W[K,F] -------------
// One wave per block; each wave computes a 16x64 output slab: 1 M-tile x 4
// N-tiles with 4 independent accumulators (no dependent WMMA->WMMA chains,
// 4x reuse of each A fragment). K is compile-time -> fully unrolled WMMA
// stream. Exact tiling: M%16==0, F%64==0, K%4==0.
// grid = (M/16, F/64), block = 32.

template <int K, bool RELU>
__global__ __launch_bounds__(32) void gemm_wmma_f32(const float* __restrict__ A,
                                                    const float* __restrict__ W,
                                                    float* __restrict__ out,
                                                    int F) {
    const int lane = threadIdx.x & 31;
    const int half = lane >> 4;                 // 0 or 1
    const int l16  = lane & 15;
    const int tm   = blockIdx.x;                // M tile
    const int cn0  = blockIdx.y * 64 + l16;     // first of 4 columns this lane owns

    const int row = tm * 16 + l16;              // A row held by this lane
    const float* __restrict__ arow = A + (size_t)row * K;

    v8f c0 = {0.f,0.f,0.f,0.f,0.f,0.f,0.f,0.f};
    v8f c1 = c0, c2 = c0, c3 = c0;

#pragma unroll
    for (int k = 0; k < K; k += 4) {
        const int ka = k + 2 * half;            // even -> 8B-aligned v2f load
        v2f a = *(const v2f*)(arow + ka);
        if (RELU) {
            a.x = fmaxf(a.x, 0.0f);
            a.y = fmaxf(a.y, 0.0f);
        }
        const float* __restrict__ w0 = W + (size_t)ka * F + cn0;        // row ka
        const float* __restrict__ w1 = w0 + F;                          // row ka+1
        v2f b0 = {w0[0],  w1[0]};
        v2f b1 = {w0[16], w1[16]};
        v2f b2 = {w0[32], w1[32]};
        v2f b3 = {w0[48], w1[48]};
        // 8 args: (neg_a, A, neg_b, B, c_mod, C, reuse_a, reuse_b)
        c0 = __builtin_amdgcn_wmma_f32_16x16x4_f32(false, a, false, b0, (short)0, c0, false, false);
        c1 = __builtin_amdgcn_wmma_f32_16x16x4_f32(false, a, false, b1, (short)0, c1, false, false);
        c2 = __builtin_amdgcn_wmma_f32_16x16x4_f32(false, a, false, b2, (short)0, c2, false, false);
        c3 = __builtin_amdgcn_wmma_f32_16x16x4_f32(false, a, false, b3, (short)0, c3, false, false);
    }

    // C/D layout: VGPR r -> M = tm*16 + r + 8*half, N = cn0 + 16*t
    const int orow = tm * 16 + 8 * half;
#pragma unroll
    for (int r = 0; r < 8; ++r) {
        float* o = out + (size_t)(orow + r) * F + cn0;
        o[0]  = c0[r];
        o[16] = c1[r];
        o[32] = c2[r];
        o[48] = c3[r];
    }
}

// ---------- edge scatter: out[dst] += h[src] * norm ----------
// chunks = F/4 = 1<<CSHIFT; thread t -> (edge = t>>CSHIFT, chunk = t&mask).
// Consecutive lanes cover one edge's features: float4 gather coalesces to
// b128 loads; scatter is 4x global_atomic_add_f32 (L2-resident RMW).

template <int CSHIFT>
__global__ void edge_scatter(const int* __restrict__ src, const int* __restrict__ dst,
                             const float* __restrict__ norm, const float* __restrict__ h,
                             float* __restrict__ out, int E) {
    const int F = 4 << CSHIFT;
    int t = blockIdx.x * blockDim.x + threadIdx.x;
    if (t >= (E << CSHIFT)) return;
    const int e = t >> CSHIFT;
    const int c = t & ((1 << CSHIFT) - 1);
    const int s = src[e];
    const int d = dst[e];
    const float nv = norm[e];
    const float4 hv = *(const float4*)(h + (size_t)s * F + c * 4);
    float* o = out + (size_t)d * F + c * 4;
    atomicAdd(o + 0, hv.x * nv);
    atomicAdd(o + 1, hv.y * nv);
    atomicAdd(o + 2, hv.z * nv);
    atomicAdd(o + 3, hv.w * nv);
}

// self-loop term: out[i,:] += h[i,:] * dinv[i]^2, float4-vectorized.
// Stream-ordered after the atomic kernel, so plain RMW is race-free.
// F4 = F/4 = 1<<F4SHIFT.
template <int F4SHIFT>
__global__ void self_loop4(const float* __restrict__ dinv, const float4* __restrict__ h,
                           float4* __restrict__ out, int total4) {
    int i = blockIdx.x * blockDim.x + threadIdx.x;
    if (i < total4) {
        const float dv = dinv[i >> F4SHIFT];
        const float s = dv * dv;
        float4 hv = h[i];
        float4 ov = out[i];
        ov.x += hv.x * s;
        ov.y += hv.y * s;
        ov.z += hv.z * s;
        ov.w += hv.w * s;
        out[i] = ov;
    }
}

// ---------------- launch ----------------

extern "C" void kernel_launch(void* const* d_in, const int* in_sizes, int n_in,
                              void* d_out, int out_size, void* d_ws, size_t ws_size,
                              hipStream_t stream) {
    const float* x  = (const float*)d_in[0];
    const int*   ei = (const int*)d_in[1];   // [2, E] row-major: row 0 = src, row 1 = dst
    const float* W1 = (const float*)d_in[2];
    const float* b1 = (const float*)d_in[3];
    const float* W2 = (const float*)d_in[4];
    const float* b2 = (const float*)d_in[5];
    const float* W3 = (const float*)d_in[6];
    const float* b3 = (const float*)d_in[7];
    float* out = (float*)d_out;

    const int* src = ei;
    const int* dst = ei + NEDGES;

    // workspace layout (256B aligned)
    char*  ws  = (char*)d_ws;
    size_t off = 0;
    auto carve = [&](size_t bytes) -> float* {
        float* p = (float*)(ws + off);
        off = (off + bytes + 255) & ~(size_t)255;
        return p;
    };
    float* dinv = carve((size_t)NNODES * 4);        //  0.2 MB
    float* norm = carve((size_t)NEDGES * 4);        //  3.2 MB
    float* h    = carve((size_t)NNODES * 128 * 4);  // 25.6 MB  (GEMM output)
    float* act  = carve((size_t)NNODES * 128 * 4);  // 25.6 MB  (aggregated activations)

    const int T = 256;
    auto blks = [](long long n, int t) { return (int)((n + t - 1) / t); };

    // ---- normalization: dinv, norm ----
    zero_f32<<<blks(NNODES, T), T, 0, stream>>>(dinv, NNODES);
    deg_count<<<blks(NEDGES, T), T, 0, stream>>>(dst, dinv, NEDGES);
    deg_to_dinv<<<blks(NNODES, T), T, 0, stream>>>(dinv, NNODES);
    edge_norm<<<blks(NEDGES, T), T, 0, stream>>>(src, dst, dinv, norm, NEDGES);

    const int MT = NNODES / 16;  // 3125 exact

    // ---- layer 1: h = x @ W1 (128->128); act = agg(h) + b1 ----
    gemm_wmma_f32<128, false><<<dim3(MT, 2), 32, 0, stream>>>(x, W1, h, 128);
    bias_init4<<<blks((long long)NNODES * 32, T), T, 0, stream>>>((float4*)act, (const float4*)b1, NNODES * 32, 31);
    edge_scatter<5><<<blks((long long)NEDGES * 32, T), T, 0, stream>>>(src, dst, norm, h, act, NEDGES);
    self_loop4<5><<<blks((long long)NNODES * 32, T), T, 0, stream>>>(dinv, (const float4*)h, (float4*)act, NNODES * 32);

    // ---- layer 2: h = relu(act) @ W2 (128->64); act = agg(h) + b2 ----
    gemm_wmma_f32<128, true><<<dim3(MT, 1), 32, 0, stream>>>(act, W2, h, 64);
    bias_init4<<<blks((long long)NNODES * 16, T), T, 0, stream>>>((float4*)act, (const float4*)b2, NNODES * 16, 15);
    edge_scatter<4><<<blks((long long)NEDGES * 16, T), T, 0, stream>>>(src, dst, norm, h, act, NEDGES);
    self_loop4<4><<<blks((long long)NNODES * 16, T), T, 0, stream>>>(dinv, (const float4*)h, (float4*)act, NNODES * 16);

    // ---- layer 3: h = relu(act) @ W3 (64->128); out = agg(h) + b3 ----
    gemm_wmma_f32<64, true><<<dim3(MT, 2), 32, 0, stream>>>(act, W3, h, 128);
    bias_init4<<<blks((long long)NNODES * 32, T), T, 0, stream>>>((float4*)out, (const float4*)b3, NNODES * 32, 31);
    edge_scatter<5><<<blks((long long)NEDGES * 32, T), T, 0, stream>>>(src, dst, norm, h, out, NEDGES);
    self_loop4<5><<<blks((long long)NNODES * 32, T), T, 0, stream>>>(dinv, (const float4*)h, (float4*)out, NNODES * 32);
}